// neural_network_60979945669198
// MI455X (gfx1250) — compile-verified
//
#include <hip/hip_runtime.h>

// ---------------------------------------------------------------------------
// J2 plasticity scan + output projection for MI455X (gfx1250, wave32)
//   Stage A: fused strain projection + sequential J2 radial-return scan
//            (1 thread / particle, 8192 particles, S=2048 dependent steps)
//   Stage B: (B*S, 192) x (192, 3->padded 16) contraction via
//            V_WMMA_F32_16X16X4_F32 (f32 WMMA, no precision loss)
// ---------------------------------------------------------------------------

typedef __attribute__((ext_vector_type(2))) float v2f;
typedef __attribute__((ext_vector_type(8))) float v8f;

// Material constants (computed in double, stored as f32 like the reference)
constexpr double dE    = 3130.0;
constexpr double dNU   = 0.37;
constexpr double dSIGY = 64.8;
constexpr double dHARD = 100.0;
constexpr double dG    = dE / (2.0 * (1.0 + dNU));
constexpr double dLAM  = dE * dNU / ((1.0 + dNU) * (1.0 - 2.0 * dNU));
constexpr double dC    = dE / ((1.0 + dNU) * (1.0 - 2.0 * dNU));

constexpr float fG     = (float)dG;
constexpr float fLAM   = (float)dLAM;
constexpr float fC11   = (float)(dC * (1.0 - dNU));
constexpr float fC12   = (float)(dC * dNU);
constexpr float fSIGY  = (float)dSIGY;
constexpr float fHARD  = (float)dHARD;
constexpr float f3G    = (float)(3.0 * dG);
constexpr float fINV3GH = (float)(1.0 / (3.0 * dG + dHARD));

#define NB   128      // batch
#define NS   2048     // sequence (scan) length
#define LS   192      // latent size (= 64 particles * 3 components per b)
#define NO   3        // output channels
#define NPADN 16      // WMMA N padding

// ---------------------------------------------------------------------------
// Kernel 0: B-panel prep.  Bpad[n][k] = n<3 ? softplus(W2[n,k]) : 0
// Stable softplus: max(w,0) + log1p(exp(-|w|))
// ---------------------------------------------------------------------------
__global__ void prep_bpanel_kernel(const float* __restrict__ W2,
                                   float* __restrict__ Bpad) {
  int k = threadIdx.x;            // 0..191
  if (k >= LS) return;
#pragma unroll
  for (int n = 0; n < NPADN; ++n) {
    float v = 0.0f;
    if (n < NO) {
      float w = W2[n * LS + k];
      v = fmaxf(w, 0.0f) + log1pf(__expf(-fabsf(w)));
    }
    Bpad[n * LS + k] = v;
  }
}

// ---------------------------------------------------------------------------
// Kernel 1: fused strain projection + J2 scan.
// grid = 128 blocks (one per b), block = 64 threads (one per particle j).
// x[b,s,:] address is wave-uniform -> scalar-cache loads.
// ---------------------------------------------------------------------------
__global__ __launch_bounds__(64) void j2_scan_kernel(
    const float* __restrict__ x,     // (B, S, 3)
    const float* __restrict__ W1,    // (192, 3)
    float* __restrict__ stress) {    // (B*S, 192) workspace
  const int b = blockIdx.x;          // 0..127
  const int j = threadIdx.x;         // 0..63 particle within b

  // W1 rows for this particle's 3 strain components (stay in VGPRs)
  const float* w = W1 + (3 * j) * 3;
  const float w00 = w[0], w01 = w[1], w02 = w[2];
  const float w10 = w[3], w11 = w[4], w12 = w[5];
  const float w20 = w[6], w21 = w[7], w22 = w[8];

  float eps0 = 0.f, eps1 = 0.f, eps2 = 0.f;
  float sig0 = 0.f, sig1 = 0.f, sig2 = 0.f;
  float szz  = 0.f, eqps = 0.f;

  const float* __restrict__ xb = x + (size_t)b * NS * 3;
  float* __restrict__ outb = stress + (size_t)b * NS * LS + 3 * j;

  for (int s = 0; s < NS; ++s) {
    // uniform per wave -> s_load path
    const float x0 = xb[3 * s + 0];
    const float x1 = xb[3 * s + 1];
    const float x2 = xb[3 * s + 2];

    // strain_t = W1[3j..3j+2,:] . x[b,s,:]
    const float st0 = w00 * x0 + w01 * x1 + w02 * x2;
    const float st1 = w10 * x0 + w11 * x1 + w12 * x2;
    const float st2 = w20 * x0 + w21 * x1 + w22 * x2;

    // elastic trial
    const float de0 = st0 - eps0;
    const float de1 = st1 - eps1;
    const float de2 = st2 - eps2;
    const float tr0 = sig0 + fC11 * de0 + fC12 * de1;
    const float tr1 = sig1 + fC12 * de0 + fC11 * de1;
    const float tr2 = sig2 + fG * de2;
    const float tzz = szz + fLAM * (de0 + de1);

    // radial return
    const float pm  = (tr0 + tr1 + tzz) * (1.0f / 3.0f);
    const float dxx = tr0 - pm;
    const float dyy = tr1 - pm;
    const float dzz = tzz - pm;
    const float dxy = tr2;
    const float qq  = 1.5f * (dxx * dxx + dyy * dyy + dzz * dzz +
                              2.0f * dxy * dxy) + 1e-12f;
    const float q   = sqrtf(qq);
    const float f   = q - (fSIGY + fHARD * eqps);
    const float dgam = (f > 0.0f) ? f * fINV3GH : 0.0f;
    // fac = 1 - 3G*dgam/q  via rcp + one Newton step (q >= 1e-6)
    float rq = __builtin_amdgcn_rcpf(q);
    rq = rq * (2.0f - q * rq);
    const float fac = 1.0f - f3G * dgam * rq;

    sig0 = pm + fac * dxx;
    sig1 = pm + fac * dyy;
    sig2 = fac * dxy;
    szz  = pm + fac * dzz;
    eqps = eqps + dgam;
    eps0 = st0; eps1 = st1; eps2 = st2;

    float* o = outb + (size_t)s * LS;
    o[0] = sig0;
    o[1] = sig1;
    o[2] = sig2;
  }
}

// ---------------------------------------------------------------------------
// Kernel 2: out (M=B*S, N=3) = stress (M, K=192) x Bpad^T (K=192, N=16pad)
// using V_WMMA_F32_16X16X4_F32.  One wave per 16-row tile, 48 WMMAs.
// A layout (16x4 f32, 2 VGPRs): lane L -> row (L&15); VGPR0/1 hold
// K = 2*(L>>4) and 2*(L>>4)+1 -> one aligned float2 load per lane per step.
// B layout mirrors A with N = (L&15).
// ---------------------------------------------------------------------------
__global__ __launch_bounds__(256) void out_gemm_wmma_kernel(
    const float* __restrict__ stress,  // (B*S, 192)
    const float* __restrict__ Bpad,    // (16, 192)
    float* __restrict__ out) {         // (B*S, 3)
  const int lane = threadIdx.x & 31;
  const int wave = threadIdx.x >> 5;
  const int tile = blockIdx.x * 8 + wave;   // 16384 tiles of 16 rows
  const int grp  = lane >> 4;               // 0 or 1
  const int ln   = lane & 15;               // row-in-tile / out column

  const float* __restrict__ arow =
      stress + ((size_t)tile * 16 + ln) * LS + 2 * grp;
  const float* __restrict__ brow = Bpad + ln * LS + 2 * grp;

  v8f acc = {};
#pragma unroll
  for (int k = 0; k < LS; k += 4) {
    v2f a = *(const v2f*)(arow + k);
    v2f bm = *(const v2f*)(brow + k);
    // (neg_a, A, neg_b, B, c_mod, C, reuse_a, reuse_b)
    acc = __builtin_amdgcn_wmma_f32_16x16x4_f32(
        false, a, false, bm, (short)0, acc, false, false);
  }

  // D layout: lane L -> N=(L&15); VGPR r -> M = r + 8*(L>>4)
  if (ln < NO) {
#pragma unroll
    for (int r = 0; r < 8; ++r) {
      const size_t m = (size_t)tile * 16 + r + 8 * grp;
      out[m * NO + ln] = acc[r];
    }
  }
}

// ---------------------------------------------------------------------------
extern "C" void kernel_launch(void* const* d_in, const int* in_sizes, int n_in,
                              void* d_out, int out_size, void* d_ws,
                              size_t ws_size, hipStream_t stream) {
  (void)in_sizes; (void)n_in; (void)out_size; (void)ws_size;
  const float* x  = (const float*)d_in[0];   // (128, 2048, 3)
  const float* W1 = (const float*)d_in[1];   // (192, 3)
  const float* W2 = (const float*)d_in[2];   // (3, 192)
  float* out = (float*)d_out;                // (128, 2048, 3)

  float* stress = (float*)d_ws;                          // 192 MB
  float* Bpad   = stress + (size_t)NB * NS * LS;         // + 12 KB

  prep_bpanel_kernel<<<1, LS, 0, stream>>>(W2, Bpad);
  j2_scan_kernel<<<NB, 64, 0, stream>>>(x, W1, stress);
  const int tiles = (NB * NS) / 16;                      // 16384
  out_gemm_wmma_kernel<<<tiles / 8, 256, 0, stream>>>(stress, Bpad, out);
}